// FakeImg_59365037965348
// MI455X (gfx1250) — compile-verified
//
#include <hip/hip_runtime.h>
#include <stdint.h>

// Problem constants (from reference)
#define HH 1024
#define WW 1024
#define PP 128           // patch edge
#define BB 512           // batch
#define ROWS_PER_BLK 8   // output rows staged per block
#define THREADS 256

// LDS staging: ROWS_PER_BLK * 128 px * 6 ch * 4B = 24 KB, laid out in final
// output-interleaved order [i][j][s*3+c] so the drain is a linear copy.
#define LDS_BYTES (ROWS_PER_BLK * PP * 6 * 4)

__global__ __launch_bounds__(THREADS) void fakeimg_gather_async(
    const float* __restrict__ img, const int* __restrict__ ys,
    const int* __restrict__ xs, float* __restrict__ out) {
  __shared__ __align__(16) float smem[ROWS_PER_BLK * PP * 6];

  const int blk = blockIdx.x;
  const int b   = blk >> 4;        // 128/ROWS_PER_BLK = 16 row-groups per batch elem
  const int ig  = blk & 15;
  const int i0  = ig * ROWS_PER_BLK;
  const int t   = threadIdx.x;

  // Patch origins for this batch element (block-uniform -> scalar loads)
  const int y0 = ys[2 * b + 0] + i0;
  const int y1 = ys[2 * b + 1] + i0;
  const int x0 = xs[2 * b + 0];
  const int x1 = xs[2 * b + 1];

  const uint64_t imgBase = (uint64_t)(uintptr_t)img;
  // Low 32 bits of the generic pointer to __shared__ == LDS byte address.
  const unsigned ldsBase = (unsigned)(uintptr_t)&smem[0];

  // ---------------- Phase 1: async gather, global -> LDS -------------------
  // 2048 pixels per block (8 rows x 2 patches x 128 cols), 8 per thread.
  // Each pixel = 3 dwords; channel offsets ride the instruction IOFFSET,
  // which applies to BOTH the memory and LDS address.
#pragma unroll
  for (int n = 0; n < (ROWS_PER_BLK * 2 * PP) / THREADS; ++n) {
    const int p   = t + THREADS * n;
    const int i   = p >> 8;        // 0..7   (row within chunk)
    const int rem = p & 255;
    const int s   = rem >> 7;      // 0..1   (which stacked patch)
    const int j   = rem & 127;     // 0..127 (column within patch)

    const int row = ((s ? y1 : y0) + i) & (HH - 1);   // torus wrap (pow2)
    const int col = ((s ? x1 : x0) + j) & (WW - 1);

    const uint64_t g = imgBase + (uint64_t)(((row << 10) + col) * 12);
    const unsigned l = ldsBase + (unsigned)(i * (PP * 6 * 4) + j * 24 + s * 12);

    asm volatile(
        "global_load_async_to_lds_b32 %0, %1, off\n\t"
        "global_load_async_to_lds_b32 %0, %1, off offset:4\n\t"
        "global_load_async_to_lds_b32 %0, %1, off offset:8"
        :: "v"(l), "v"(g) : "memory");
  }

  // All async LDS writes done, then make them visible block-wide.
  asm volatile("s_wait_asynccnt 0x0" ::: "memory");
  __syncthreads();

  // ---------------- Phase 2: async drain, LDS -> global (NT stores) -------
  // 24 KB contiguous chunk per block; 6 x B128 per thread, stride 4096 B.
  const uint64_t outBase =
      (uint64_t)(uintptr_t)out +
      ((uint64_t)b * PP + (uint64_t)i0) * (uint64_t)(PP * 6 * 4);
  const uint64_t go = outBase + (uint64_t)(t * 16);
  const unsigned lo = ldsBase + (unsigned)(t * 16);

  asm volatile(
      "global_store_async_from_lds_b128 %0, %1, off th:TH_STORE_NT\n\t"
      "global_store_async_from_lds_b128 %0, %1, off offset:4096 th:TH_STORE_NT\n\t"
      "global_store_async_from_lds_b128 %0, %1, off offset:8192 th:TH_STORE_NT\n\t"
      "global_store_async_from_lds_b128 %0, %1, off offset:12288 th:TH_STORE_NT\n\t"
      "global_store_async_from_lds_b128 %0, %1, off offset:16384 th:TH_STORE_NT\n\t"
      "global_store_async_from_lds_b128 %0, %1, off offset:20480 th:TH_STORE_NT"
      :: "v"(go), "v"(lo) : "memory");

  // Don't leave the workgroup (and its LDS) with outstanding async ops.
  asm volatile("s_wait_asynccnt 0x0" ::: "memory");
}

extern "C" void kernel_launch(void* const* d_in, const int* in_sizes, int n_in,
                              void* d_out, int out_size, void* d_ws, size_t ws_size,
                              hipStream_t stream) {
  (void)in_sizes; (void)n_in; (void)out_size; (void)d_ws; (void)ws_size;
  const float* img = (const float*)d_in[0];
  const int*   ysp = (const int*)d_in[1];
  const int*   xsp = (const int*)d_in[2];
  float*       out = (float*)d_out;

  const int blocks = BB * (PP / ROWS_PER_BLK);  // 512 * 16 = 8192
  hipLaunchKernelGGL(fakeimg_gather_async, dim3(blocks), dim3(THREADS), 0, stream,
                     img, ysp, xsp, out);
}